// HierarchicalGraphAttention_85856396247186
// MI455X (gfx1250) — compile-verified
//
#include <hip/hip_runtime.h>

#define N_NODES 50000
#define N_EDGES 800000
#define FEAT    128
#define TOTE    (N_EDGES + N_NODES)
#define NGRAPH  8
#define MTILES  (N_NODES / 16)   // 3125, exact
#define NEG_SLOPE 0.2f

typedef float v2f __attribute__((ext_vector_type(2)));
typedef float v8f __attribute__((ext_vector_type(8)));

// ---------------------------------------------------------------- WMMA helper
__device__ __forceinline__ v8f wmma4(v2f a, v2f b, v8f c) {
    // V_WMMA_F32_16X16X4_F32 : D = A(16x4) * B(4x16) + C(16x16), all fp32
    return __builtin_amdgcn_wmma_f32_16x16x4_f32(
        /*neg_a=*/false, a, /*neg_b=*/false, b,
        /*c_mod=*/(short)0, c, /*reuse_a=*/false, /*reuse_b=*/false);
}

__device__ __forceinline__ float atomAddF(float* p, float v) {
    return __hip_atomic_fetch_add(p, v, __ATOMIC_RELAXED, __HIP_MEMORY_SCOPE_AGENT);
}

// monotone float <-> uint key for atomic max on signed floats
__device__ __forceinline__ unsigned f2key(float f) {
    unsigned b = __float_as_uint(f);
    return (b & 0x80000000u) ? ~b : (b | 0x80000000u);
}
__device__ __forceinline__ float key2f(unsigned k) {
    return (k & 0x80000000u) ? __uint_as_float(k & 0x7fffffffu)
                             : __uint_as_float(~k);
}

// ------------------------------------------------ generic WMMA GEMM (fp32)
// C[M,ncols] = A[M,K] @ B[K,ncols] (+bias) (optional relu). Wave = 16x64 tile.
__global__ void __launch_bounds__(256) wmma_gemm_kernel(
    const float* __restrict__ A, const float* __restrict__ Bm,
    const float* __restrict__ bias, float* __restrict__ C,
    int mtiles, int K, int ncols, int relu) {
    const int lane = threadIdx.x & 31;
    const int wid  = threadIdx.x >> 5;
    const int t    = blockIdx.x * 8 + wid;
    if (t >= mtiles) return;
    const int nbase = blockIdx.y * 64;
    const int kb    = (lane >> 4) << 1;            // 0 or 2: lane-half K pair
    const float* arow = A + (size_t)(t * 16 + (lane & 15)) * K;
    const float* bcol = Bm + nbase + (lane & 15);
    v8f c0 = {}, c1 = {}, c2 = {}, c3 = {};
    #pragma unroll 4
    for (int k = 0; k < K; k += 4) {
        v2f a;  a.x = arow[k + kb];  a.y = arow[k + kb + 1];
        const float* bp = bcol + (size_t)(k + kb) * ncols;
        v2f b0; b0.x = bp[0];  b0.y = bp[ncols];
        v2f b1; b1.x = bp[16]; b1.y = bp[ncols + 16];
        v2f b2; b2.x = bp[32]; b2.y = bp[ncols + 32];
        v2f b3; b3.x = bp[48]; b3.y = bp[ncols + 48];
        c0 = wmma4(a, b0, c0);
        c1 = wmma4(a, b1, c1);
        c2 = wmma4(a, b2, c2);
        c3 = wmma4(a, b3, c3);
    }
    const int col0  = nbase + (lane & 15);
    const int rbase = t * 16 + ((lane >> 4) << 3);
    #pragma unroll
    for (int r = 0; r < 8; ++r) {
        float* crow = C + (size_t)(rbase + r) * ncols;
        float v0 = c0[r], v1 = c1[r], v2 = c2[r], v3 = c3[r];
        if (bias) {
            v0 += bias[col0];      v1 += bias[col0 + 16];
            v2 += bias[col0 + 32]; v3 += bias[col0 + 48];
        }
        if (relu) {
            v0 = fmaxf(v0, 0.f); v1 = fmaxf(v1, 0.f);
            v2 = fmaxf(v2, 0.f); v3 = fmaxf(v3, 0.f);
        }
        crow[col0]      = v0; crow[col0 + 16] = v1;
        crow[col0 + 32] = v2; crow[col0 + 48] = v3;
    }
}

// ------------- fusion GEMM1: A = [local | regional | gfeat[batch]] (K=384)
// hmid[N,256] = relu(cat @ fu_W1 + fu_b1), cat never materialized.
__global__ void __launch_bounds__(256) wmma_fuse1_kernel(
    const float* __restrict__ localb, const float* __restrict__ regionalb,
    const float* __restrict__ gfeat, const int* __restrict__ batch,
    const float* __restrict__ W, const float* __restrict__ bias,
    float* __restrict__ C, int mtiles) {
    const int lane = threadIdx.x & 31;
    const int wid  = threadIdx.x >> 5;
    const int t    = blockIdx.x * 8 + wid;
    if (t >= mtiles) return;
    const int nbase = blockIdx.y * 64;            // ncols = 256
    const int kb    = (lane >> 4) << 1;
    const int node  = t * 16 + (lane & 15);
    const int bg    = batch[node];
    const float* s0 = localb    + (size_t)node * FEAT;
    const float* s1 = regionalb + (size_t)node * FEAT;
    const float* s2 = gfeat     + (size_t)bg   * FEAT;
    const float* bcol = W + nbase + (lane & 15);
    v8f c0 = {}, c1 = {}, c2 = {}, c3 = {};
    #pragma unroll 4
    for (int k = 0; k < 384; k += 4) {
        const int ka  = k + kb;                   // region uniform over wave
        const int reg = ka >> 7;
        const int col = ka & 127;
        const float* sp = (reg == 0) ? s0 : ((reg == 1) ? s1 : s2);
        v2f a;  a.x = sp[col]; a.y = sp[col + 1];
        const float* bp = bcol + (size_t)ka * 256;
        v2f b0; b0.x = bp[0];  b0.y = bp[256];
        v2f b1; b1.x = bp[16]; b1.y = bp[256 + 16];
        v2f b2; b2.x = bp[32]; b2.y = bp[256 + 32];
        v2f b3; b3.x = bp[48]; b3.y = bp[256 + 48];
        c0 = wmma4(a, b0, c0);
        c1 = wmma4(a, b1, c1);
        c2 = wmma4(a, b2, c2);
        c3 = wmma4(a, b3, c3);
    }
    const int col0  = nbase + (lane & 15);
    const int rbase = t * 16 + ((lane >> 4) << 3);
    #pragma unroll
    for (int r = 0; r < 8; ++r) {
        float* crow = C + (size_t)(rbase + r) * 256;
        crow[col0]      = fmaxf(c0[r] + bias[col0],      0.f);
        crow[col0 + 16] = fmaxf(c1[r] + bias[col0 + 16], 0.f);
        crow[col0 + 32] = fmaxf(c2[r] + bias[col0 + 32], 0.f);
        crow[col0 + 48] = fmaxf(c3[r] + bias[col0 + 48], 0.f);
    }
}

// -------------------------------------------------- per-graph mean pooling
__global__ void __launch_bounds__(128) pool_sum_kernel(
    const float* __restrict__ x, const int* __restrict__ batch,
    float* __restrict__ gsum, float* __restrict__ gcnt) {
    __shared__ float lsum[NGRAPH * FEAT];
    __shared__ float lcnt[NGRAPH];
    const int f = threadIdx.x;
    #pragma unroll
    for (int b = 0; b < NGRAPH; ++b) lsum[b * FEAT + f] = 0.f;
    if (f < NGRAPH) lcnt[f] = 0.f;
    __syncthreads();
    const int base = blockIdx.x * 64;
    for (int i = 0; i < 64; ++i) {
        const int n = base + i;
        if (n >= N_NODES) break;
        const int b = batch[n];
        lsum[b * FEAT + f] += x[(size_t)n * FEAT + f];  // thread owns column f
        if (f == 0) lcnt[b] += 1.f;
    }
    __syncthreads();
    #pragma unroll
    for (int b = 0; b < NGRAPH; ++b) atomAddF(&gsum[b * FEAT + f], lsum[b * FEAT + f]);
    if (f < NGRAPH) atomAddF(&gcnt[f], lcnt[f]);
}

// gfeat_g = relu(mean_g @ W1 + b1) @ W2 + b2  (B=8 rows only)
__global__ void __launch_bounds__(128) gp_mlp_kernel(
    const float* __restrict__ gsum, const float* __restrict__ gcnt,
    const float* __restrict__ W1, const float* __restrict__ b1,
    const float* __restrict__ W2, const float* __restrict__ b2,
    float* __restrict__ gfeat) {
    __shared__ float mean[FEAT], t1[FEAT];
    const int g = blockIdx.x, f = threadIdx.x;
    mean[f] = gsum[g * FEAT + f] / gcnt[g];
    __syncthreads();
    float a = b1[f];
    for (int k = 0; k < FEAT; ++k) a += mean[k] * W1[k * FEAT + f];
    t1[f] = fmaxf(a, 0.f);
    __syncthreads();
    float a2 = b2[f];
    for (int k = 0; k < FEAT; ++k) a2 += t1[k] * W2[k * FEAT + f];
    gfeat[g * FEAT + f] = a2;
}

// ------------------------------------- per-node attention alpha terms (wave32)
__global__ void __launch_bounds__(256) alpha_kernel(
    const float* __restrict__ h, const float* __restrict__ a_src,
    const float* __restrict__ a_dst, float* __restrict__ al_s,
    float* __restrict__ al_d) {
    const int lane = threadIdx.x & 31, wid = threadIdx.x >> 5;
    const int n = blockIdx.x * 8 + wid;
    if (n >= N_NODES) return;
    const float4 hv = *(const float4*)(h + (size_t)n * FEAT + lane * 4);
    const float4 as = *(const float4*)(a_src + lane * 4);
    const float4 ad = *(const float4*)(a_dst + lane * 4);
    float ps = hv.x * as.x + hv.y * as.y + hv.z * as.z + hv.w * as.w;
    float pd = hv.x * ad.x + hv.y * ad.y + hv.z * ad.z + hv.w * ad.w;
    #pragma unroll
    for (int m = 1; m < 8; m <<= 1) {
        ps += __shfl_xor(ps, m, 32);
        pd += __shfl_xor(pd, m, 32);
    }
    if ((lane & 7) == 0) {
        const int head = lane >> 3;
        al_s[n * 4 + head] = ps;
        al_d[n * 4 + head] = pd;
    }
}

// --------------------------------------------- edge softmax: pass 1 (max)
__global__ void __launch_bounds__(256) edge_logit_kernel(
    const int* __restrict__ src, const int* __restrict__ dst,
    const float* __restrict__ al_s, const float* __restrict__ al_d,
    float* __restrict__ exbuf, unsigned* __restrict__ maxkey) {
    const int e = blockIdx.x * blockDim.x + threadIdx.x;
    if (e >= TOTE) return;
    int s, d;
    if (e < N_EDGES) { s = src[e]; d = dst[e]; } else { s = d = e - N_EDGES; }
    #pragma unroll
    for (int hh = 0; hh < 4; ++hh) {
        float l = al_s[s * 4 + hh] + al_d[d * 4 + hh];
        l = (l >= 0.f) ? l : NEG_SLOPE * l;
        exbuf[(size_t)e * 4 + hh] = l;
        atomicMax(&maxkey[d * 4 + hh], f2key(l));
    }
}

// --------------------------------------------- edge softmax: pass 2 (exp+sum)
__global__ void __launch_bounds__(256) edge_exp_kernel(
    const int* __restrict__ dst, float* __restrict__ exbuf,
    const unsigned* __restrict__ maxkey, float* __restrict__ denom) {
    const int e = blockIdx.x * blockDim.x + threadIdx.x;
    if (e >= TOTE) return;
    const int d = (e < N_EDGES) ? dst[e] : (e - N_EDGES);
    #pragma unroll
    for (int hh = 0; hh < 4; ++hh) {
        const float m  = key2f(maxkey[d * 4 + hh]);
        const float ex = __expf(exbuf[(size_t)e * 4 + hh] - m);
        exbuf[(size_t)e * 4 + hh] = ex;
        atomAddF(&denom[d * 4 + hh], ex);
    }
}

// ------------------------------ edge message scatter: wave per edge, fp32 atomics
__global__ void __launch_bounds__(256) edge_scatter_kernel(
    const int* __restrict__ src, const int* __restrict__ dst,
    const float* __restrict__ h, const float* __restrict__ exbuf,
    const float* __restrict__ denom, float* __restrict__ acc) {
    const int lane = threadIdx.x & 31, wid = threadIdx.x >> 5;
    const int e = blockIdx.x * 8 + wid;
    if (e >= TOTE) return;
    int s, d;
    if (e < N_EDGES) { s = src[e]; d = dst[e]; } else { s = d = e - N_EDGES; }
    const int hh = lane >> 3;
    const float alpha = exbuf[(size_t)e * 4 + hh] / denom[d * 4 + hh];
    const float4 hv = *(const float4*)(h + (size_t)s * FEAT + lane * 4);
    float* out = acc + (size_t)d * FEAT + lane * 4;
    atomAddF(out + 0, hv.x * alpha);
    atomAddF(out + 1, hv.y * alpha);
    atomAddF(out + 2, hv.z * alpha);
    atomAddF(out + 3, hv.w * alpha);
}

// ----------------- LayerNorm (optional +bias, +residual); wave per node row
__global__ void __launch_bounds__(256) ln_kernel(
    const float* __restrict__ in, const float* __restrict__ res,
    const float* __restrict__ bias, const float* __restrict__ g,
    const float* __restrict__ b, float* __restrict__ out) {
    const int lane = threadIdx.x & 31, wid = threadIdx.x >> 5;
    const int n = blockIdx.x * 8 + wid;
    if (n >= N_NODES) return;
    const int c = lane * 4;
    float4 v = *(const float4*)(in + (size_t)n * FEAT + c);
    if (bias) {
        const float4 bb = *(const float4*)(bias + c);
        v.x += bb.x; v.y += bb.y; v.z += bb.z; v.w += bb.w;
    }
    if (res) {
        const float4 rr = *(const float4*)(res + (size_t)n * FEAT + c);
        v.x += rr.x; v.y += rr.y; v.z += rr.z; v.w += rr.w;
    }
    float s  = v.x + v.y + v.z + v.w;
    float s2 = v.x * v.x + v.y * v.y + v.z * v.z + v.w * v.w;
    #pragma unroll
    for (int m = 16; m >= 1; m >>= 1) {
        s  += __shfl_xor(s,  m, 32);
        s2 += __shfl_xor(s2, m, 32);
    }
    const float mu  = s * (1.0f / FEAT);
    const float var = s2 * (1.0f / FEAT) - mu * mu;
    const float rs  = rsqrtf(var + 1e-5f);
    const float4 gg = *(const float4*)(g + c);
    const float4 bb = *(const float4*)(b + c);
    float4 o;
    o.x = (v.x - mu) * rs * gg.x + bb.x;
    o.y = (v.y - mu) * rs * gg.y + bb.y;
    o.z = (v.z - mu) * rs * gg.z + bb.z;
    o.w = (v.w - mu) * rs * gg.w + bb.w;
    *(float4*)(out + (size_t)n * FEAT + c) = o;
}

// ---------------------------------------------------------------- launcher
extern "C" void kernel_launch(void* const* d_in, const int* in_sizes, int n_in,
                              void* d_out, int out_size, void* d_ws, size_t ws_size,
                              hipStream_t stream) {
    const float* x        = (const float*)d_in[0];
    const int*   eidx     = (const int*)d_in[1];
    const int*   src      = eidx;             // edge_index[0,:]
    const int*   dst      = eidx + N_EDGES;   // edge_index[1,:]
    const int*   batch    = (const int*)d_in[2];
    const float* loc_W    = (const float*)d_in[3];
    const float* loc_as   = (const float*)d_in[4];
    const float* loc_ad   = (const float*)d_in[5];
    const float* loc_bias = (const float*)d_in[6];
    const float* loc_g    = (const float*)d_in[7];
    const float* loc_b    = (const float*)d_in[8];
    const float* reg_W    = (const float*)d_in[9];
    const float* reg_as   = (const float*)d_in[10];
    const float* reg_ad   = (const float*)d_in[11];
    const float* reg_bias = (const float*)d_in[12];
    const float* reg_g    = (const float*)d_in[13];
    const float* reg_b    = (const float*)d_in[14];
    const float* gp_W1    = (const float*)d_in[15];
    const float* gp_b1    = (const float*)d_in[16];
    const float* gp_W2    = (const float*)d_in[17];
    const float* gp_b2    = (const float*)d_in[18];
    const float* fu_W1    = (const float*)d_in[19];
    const float* fu_b1    = (const float*)d_in[20];
    const float* fu_W2    = (const float*)d_in[21];
    const float* fu_b2    = (const float*)d_in[22];
    const float* fu_ln_g  = (const float*)d_in[23];
    const float* fu_ln_b  = (const float*)d_in[24];
    float* outp = (float*)d_out;

    // workspace carve-up (all fp32-sized slots, 16B-aligned sub-sizes)
    float* ws = (float*)d_ws;
    size_t off = 0;
    float*    h         = ws + off; off += (size_t)N_NODES * FEAT;   // 6.4M
    float*    acc       = ws + off; off += (size_t)N_NODES * FEAT;   // 6.4M
    float*    localb    = ws + off; off += (size_t)N_NODES * FEAT;   // 6.4M
    float*    regionalb = ws + off; off += (size_t)N_NODES * FEAT;   // 6.4M
    float*    al_s      = ws + off; off += (size_t)N_NODES * 4;
    float*    al_d      = ws + off; off += (size_t)N_NODES * 4;
    unsigned* maxkey    = (unsigned*)(ws + off); off += (size_t)N_NODES * 4;
    float*    denom     = ws + off; off += (size_t)N_NODES * 4;
    float*    exbuf     = ws + off; off += (size_t)TOTE * 4;         // 3.4M
    float*    hmid      = ws + off; off += (size_t)N_NODES * 256;    // 12.8M
    float*    gsum      = ws + off; off += NGRAPH * FEAT;
    float*    gcnt      = ws + off; off += NGRAPH * 8;               // padded
    float*    gfeat     = ws + off; off += NGRAPH * FEAT;

    const dim3 blk(256);
    const int  gmtBlocks = (MTILES + 7) / 8;     // 391

    // --- global context path (tiny) ---
    hipMemsetAsync(gsum, 0, NGRAPH * FEAT * sizeof(float), stream);
    hipMemsetAsync(gcnt, 0, NGRAPH * sizeof(float), stream);
    pool_sum_kernel<<<(N_NODES + 63) / 64, 128, 0, stream>>>(x, batch, gsum, gcnt);
    gp_mlp_kernel<<<NGRAPH, FEAT, 0, stream>>>(gsum, gcnt, gp_W1, gp_b1, gp_W2, gp_b2, gfeat);

    // --- two GAT blocks, sequential, sharing scratch ---
    struct Gat { const float *W, *as, *ad, *bias, *g, *b; float* out; };
    const Gat gats[2] = {
        { loc_W, loc_as, loc_ad, loc_bias, loc_g, loc_b, localb },
        { reg_W, reg_as, reg_ad, reg_bias, reg_g, reg_b, regionalb },
    };
    for (int p = 0; p < 2; ++p) {
        const Gat& G = gats[p];
        // h = x @ W  (WMMA fp32)
        wmma_gemm_kernel<<<dim3(gmtBlocks, FEAT / 64), blk, 0, stream>>>(
            x, G.W, nullptr, h, MTILES, FEAT, FEAT, 0);
        alpha_kernel<<<(N_NODES + 7) / 8, blk, 0, stream>>>(h, G.as, G.ad, al_s, al_d);
        hipMemsetAsync(maxkey, 0, (size_t)N_NODES * 4 * sizeof(unsigned), stream);
        hipMemsetAsync(denom, 0, (size_t)N_NODES * 4 * sizeof(float), stream);
        hipMemsetAsync(acc, 0, (size_t)N_NODES * FEAT * sizeof(float), stream);
        edge_logit_kernel<<<(TOTE + 255) / 256, blk, 0, stream>>>(src, dst, al_s, al_d, exbuf, maxkey);
        edge_exp_kernel<<<(TOTE + 255) / 256, blk, 0, stream>>>(dst, exbuf, maxkey, denom);
        edge_scatter_kernel<<<(TOTE + 7) / 8, blk, 0, stream>>>(src, dst, h, exbuf, denom, acc);
        // out = LN(acc + bias + x)
        ln_kernel<<<(N_NODES + 7) / 8, blk, 0, stream>>>(acc, x, G.bias, G.g, G.b, G.out);
    }

    // --- fusion MLP: hmid = relu([local|regional|gfeat[batch]] @ W1 + b1) ---
    wmma_fuse1_kernel<<<dim3(gmtBlocks, 4), blk, 0, stream>>>(
        localb, regionalb, gfeat, batch, fu_W1, fu_b1, hmid, MTILES);
    // fused = hmid @ W2 + b2   (pre-LN, straight into d_out)
    wmma_gemm_kernel<<<dim3(gmtBlocks, 2), blk, 0, stream>>>(
        hmid, fu_W2, fu_b2, outp, MTILES, 2 * FEAT, FEAT, 0);
    // final LayerNorm in place
    ln_kernel<<<(N_NODES + 7) / 8, blk, 0, stream>>>(
        outp, nullptr, nullptr, fu_ln_g, fu_ln_b, outp);
}